// Conv2d_STN_15762529976346
// MI455X (gfx1250) — compile-verified
//
#include <hip/hip_runtime.h>

typedef __attribute__((ext_vector_type(2))) float v2f;
typedef __attribute__((ext_vector_type(8))) float v8f;

// Problem constants (from setup_inputs)
static constexpr int BB   = 8;     // batch
static constexpr int CIN  = 128;   // channels
static constexpr int HH   = 56;
static constexpr int WW   = 56;
static constexpr int HO   = 54;
static constexpr int WO   = 54;
static constexpr int KD   = 1152;  // C * 3 * 3
static constexpr int NTOT = BB * HO * WO;  // 23328 (divisible by 16)
static constexpr int OUTC = 256;
static constexpr int SROW = 1156;  // K3 LDS row stride (1156 % 64 = 4 -> bank spread)
static constexpr int AROW = 60;    // K1 LDS per-ci stride: 3 rows x 20 cols

__device__ __forceinline__ v8f wmma_f32(v2f a, v2f b, v8f c) {
  // V_WMMA_F32_16X16X4_F32 : D(16x16,f32) = A(16x4,f32) * B(4x16,f32) + C
  return __builtin_amdgcn_wmma_f32_16x16x4_f32(
      /*neg_a=*/false, a, /*neg_b=*/false, b,
      /*c_mod=*/(short)0, c, /*reuse_a=*/false, /*reuse_b=*/false);
}

// ---------------------------------------------------------------------------
// K1: h = relu(conv3x3(x, w0, pad=1) + b0) as implicit GEMM via f32 WMMA.
// Block (256 thr = 8 waves) owns one (b, y, xtile): stages the zero-padded
// 128ci x 3row x 18col input patch into LDS with CDNA5 async global->LDS
// copies, then each wave computes a 16(spatial) x 16(co) tile.
// K-order is (r,s)-major / ci-minor  =>  no integer division in the hot loop.
// ---------------------------------------------------------------------------
__global__ void __launch_bounds__(256) conv0_kernel(
    const float* __restrict__ x, const float* __restrict__ w0,
    const float* __restrict__ b0, float* __restrict__ h) {
  __shared__ float As[CIN * AROW];        // 30,720 B
  const int tid = threadIdx.x;
  const int xtile = blockIdx.x;           // 4 tiles cover W=56
  const int y  = blockIdx.y;
  const int b  = blockIdx.z;
  const int x0 = xtile * 16;
  const float* xb = x + b * (CIN * HH * WW);

  // zero-fill (gives free zero padding for OOB taps)
  for (int i = tid; i < CIN * AROW; i += 256) As[i] = 0.0f;
  __syncthreads();

  // async-stage the valid patch elements: cols x0-1 .. x0+16 (18), rows y-1..y+1
  for (int i = tid; i < CIN * 3 * 18; i += 256) {
    const int ci  = i / 54;
    const int rem = i - ci * 54;
    const int r   = rem / 18;
    const int col = rem - r * 18;
    const int gy  = y + r - 1;
    const int gx  = x0 - 1 + col;
    if (gy >= 0 && gy < HH && gx >= 0 && gx < WW) {
      const unsigned lds = (unsigned)(uintptr_t)(&As[ci * AROW + r * 20 + col]);
      const float* src = xb + (ci * HH + gy) * WW + gx;
      asm volatile("global_load_async_to_lds_b32 %0, %1, off"
                   :: "v"(lds), "v"(src) : "memory");
    }
  }
  asm volatile("s_wait_asynccnt 0x0" ::: "memory");
  __syncthreads();

  // GEMM: M=16 spatial (x0..x0+15), N=16 co per wave, K = 9*128
  const int wave = tid >> 5;
  const int lane = tid & 31;
  const int half = lane >> 4;
  const int l15  = lane & 15;
  const int co   = wave * 16 + l15;       // B-matrix col N
  const float* wrow = w0 + co * KD;       // w0[co, ci, r, s] flat = co*1152 + ci*9 + rs

  v8f acc = {};
#pragma unroll
  for (int rs = 0; rs < 9; ++rs) {
    const int r = rs / 3;
    const int s = rs - r * 3;
    const float* arow = &As[r * 20 + l15 + s];   // + ci*AROW per K element
    const float* wcol = wrow + rs;               // + ci*9    per K element
#pragma unroll 2
    for (int ci0 = 0; ci0 < CIN; ci0 += 4) {
      const int cb = ci0 + half * 2;             // this lane's K base (K = ci)
      v2f a;
      a[0] = arow[cb * AROW];
      a[1] = arow[cb * AROW + AROW];
      v2f bb;
      bb[0] = wcol[cb * 9];
      bb[1] = wcol[cb * 9 + 9];
      acc = wmma_f32(a, bb, acc);
    }
  }

  const float bias = b0[co];
#pragma unroll
  for (int r = 0; r < 8; ++r) {
    const int m  = half * 8 + r;          // D layout: M = 8*(lane/16) + vgpr
    const int xo = x0 + m;
    if (xo < WW) {
      float v = acc[r] + bias;
      h[((b * CIN + co) * HH + y) * WW + xo] = v > 0.0f ? v : 0.0f;
    }
  }
}

// ---------------------------------------------------------------------------
// K2: out6 = conv3x3(h, w1, pad=0)*(1-check) + ident6 ; also identm = I - theta
// Only 6 output channels -> scalar kernel (2% of total FLOPs).
// ---------------------------------------------------------------------------
__global__ void conv1_kernel(const float* __restrict__ h,
                             const float* __restrict__ w1,
                             const int* __restrict__ checkp,
                             float* __restrict__ out6,
                             float* __restrict__ identm) {
  const int idx = blockIdx.x * blockDim.x + threadIdx.x;
  const int TOT = BB * 6 * HO * WO;
  if (idx >= TOT) return;
  int b   = idx / (6 * HO * WO);
  int rem = idx - b * (6 * HO * WO);
  int o   = rem / (HO * WO);
  int pq  = rem - o * (HO * WO);
  int ho  = pq / WO;
  int wo  = pq - ho * WO;

  const float* wr = w1 + o * KD;
  const float* hb = h + b * (CIN * HH * WW);
  float acc = 0.0f;
  for (int ci = 0; ci < CIN; ++ci) {
#pragma unroll
    for (int r = 0; r < 3; ++r) {
      const float* hp = hb + (ci * HH + ho + r) * WW + wo;
      const float* wp = wr + (ci * 3 + r) * 3;
      acc += hp[0] * wp[0] + hp[1] * wp[1] + hp[2] * wp[2];
    }
  }
  const float checkf = (float)(*checkp);
  const float I6 = (o == 0 || o == 4) ? 1.0f : 0.0f;
  const float val = acc * (1.0f - checkf) + I6;
  out6[idx] = val;                                   // (B,6,Ho,Wo) flat == idx
  const int n = (b * WO + wo) * HO + ho;             // theta index ordering
  identm[n * 6 + o] = I6 - val;                      // ident23 - theta
}

// ---------------------------------------------------------------------------
// K3: fused affine-grid + bilinear sampling (into LDS) + (16 x 256, K=1152)
// GEMM with f32 WMMA + bias + transposed score scatter. Also writes auxx/auxy.
// 256 threads = 8 waves per block; block owns 16 consecutive n.
// ---------------------------------------------------------------------------
__device__ __forceinline__ float tap(const float* __restrict__ img, int yi, int xi) {
  if (xi < 0 || xi >= WW || yi < 0 || yi >= HH) return 0.0f;
  return img[yi * WW + xi];
}

__global__ void __launch_bounds__(256) sample_gemm_kernel(
    const float* __restrict__ x, const float* __restrict__ out6,
    const float* __restrict__ lw, const float* __restrict__ lb,
    float* __restrict__ score, float* __restrict__ auxx,
    float* __restrict__ auxy) {
  extern __shared__ char smem_raw[];
  float* S   = (float*)smem_raw;              // [16][SROW] sampled im2col tile
  float* swx = S + 16 * SROW;                 // [144] bilinear wx
  float* swy = swx + 144;                     // [144] bilinear wy
  int*   sx0 = (int*)(swy + 144);             // [144] floor x
  int*   sy0 = sx0 + 144;                     // [144] floor y
  int*   sbv = sy0 + 144;                     // [16]  batch per m

  const int tid = threadIdx.x;
  const int n0  = blockIdx.x * 16;

  // ---- phase 1a: per-(m,pos) affine grid -> sampling params; write aux out
  if (tid < 144) {
    const int m   = tid / 9;
    const int pos = tid - m * 9;
    const int i   = pos / 3;
    const int j   = pos - i * 3;
    const int n   = n0 + m;
    const int b   = n / (HO * WO);
    const int r2  = n - b * (HO * WO);
    const int p   = r2 / WO;                  // (n/54)%54
    const int q   = r2 - p * WO;              // n%54
    float th[6];
#pragma unroll
    for (int c = 0; c < 6; ++c)
      th[c] = out6[((b * 6 + c) * HO + q) * WO + p];  // theta[n] = out[b,:,ho,wo]
    const float bs0 = -2.0f / 3.0f, bs2 = 2.0f / 3.0f;
    const float bj = (j == 0) ? bs0 : (j == 2 ? bs2 : 0.0f);
    const float bi = (i == 0) ? bs0 : (i == 2 ? bs2 : 0.0f);
    const float g0 = th[0] * bj + th[1] * bi + th[2];
    const float g1 = th[3] * bj + th[4] * bi + th[5];
    const float ax = (g0 + 1.0f + (float)p) * (2.0f / 55.0f) - 1.0f;
    const float ay = (g1 + 1.0f + (float)q) * (2.0f / 55.0f) - 1.0f;
    auxx[n * 9 + pos] = ax;
    auxy[n * 9 + pos] = ay;
    const float gx = (ax + 1.0f) * (0.5f * (float)WW) - 0.5f;
    const float gy = (ay + 1.0f) * (0.5f * (float)HH) - 0.5f;
    const float fx = floorf(gx), fy = floorf(gy);
    swx[tid] = gx - fx;
    swy[tid] = gy - fy;
    sx0[tid] = (int)fx;
    sy0[tid] = (int)fy;
    if (pos == 0) sbv[m] = b;
  }
  __syncthreads();

  // ---- phase 1b: bilinear gathers into LDS S tile
  for (int w = tid; w < 16 * 9 * CIN; w += 256) {
    const int c  = w & (CIN - 1);
    const int mp = w >> 7;                    // (m*9 + pos)
    const int m  = mp / 9;
    const int pos = mp - m * 9;
    const int x0i = sx0[mp], y0i = sy0[mp];
    const float wx = swx[mp], wy = swy[mp];
    const float* img = x + (sbv[m] * CIN + c) * (HH * WW);
    const float v00 = tap(img, y0i, x0i);
    const float v01 = tap(img, y0i, x0i + 1);
    const float v10 = tap(img, y0i + 1, x0i);
    const float v11 = tap(img, y0i + 1, x0i + 1);
    const float val = v00 * (1.0f - wx) * (1.0f - wy) + v01 * wx * (1.0f - wy) +
                      v10 * (1.0f - wx) * wy + v11 * wx * wy;
    S[m * SROW + c * 9 + pos] = val;
  }
  __syncthreads();

  // ---- phase 2: WMMA GEMM, 16 rows x 256 cols, K = 1152
  const int wave = tid >> 5;
  const int lane = tid & 31;
  const int half = lane >> 4;
  const int l15  = lane & 15;
  const int o0   = wave * 32 + l15;
  const int o1   = o0 + 16;
  const float* lw0 = lw + o0 * KD;
  const float* lw1 = lw + o1 * KD;
  const float* srow = S + l15 * SROW;

  v8f acc0 = {}, acc1 = {};
#pragma unroll 4
  for (int k0 = 0; k0 < KD; k0 += 4) {
    const int kb = k0 + half * 2;             // 8-byte aligned (kb even)
    const v2f a   = *(const v2f*)(srow + kb);     // ds_load_b64
    const v2f b0v = *(const v2f*)(lw0 + kb);      // global_load_b64
    const v2f b1v = *(const v2f*)(lw1 + kb);
    __builtin_prefetch(lw0 + kb + 256, 0, 1);     // global_prefetch_b8 (spec ok OOB)
    __builtin_prefetch(lw1 + kb + 256, 0, 1);
    acc0 = wmma_f32(a, b0v, acc0);
    acc1 = wmma_f32(a, b1v, acc1);
  }

  const float bias0 = lb[o0];
  const float bias1 = lb[o1];
#pragma unroll
  for (int r = 0; r < 8; ++r) {
    const int m  = half * 8 + r;
    const int n  = n0 + m;
    const int b  = n / (HO * WO);
    const int r2 = n - b * (HO * WO);
    const int p  = r2 / WO;
    const int q  = r2 - p * WO;
    // score (B, O, Wo, Ho): score[b, o, q, p]
    score[((b * OUTC + o0) * WO + q) * HO + p] = acc0[r] + bias0;
    score[((b * OUTC + o1) * WO + q) * HO + p] = acc1[r] + bias1;
  }
}

// ---------------------------------------------------------------------------
extern "C" void kernel_launch(void* const* d_in, const int* in_sizes, int n_in,
                              void* d_out, int out_size, void* d_ws, size_t ws_size,
                              hipStream_t stream) {
  const float* x     = (const float*)d_in[0];
  const float* w0    = (const float*)d_in[1];
  const float* b0    = (const float*)d_in[2];
  const float* w1    = (const float*)d_in[3];
  const float* lw    = (const float*)d_in[4];
  const float* lb    = (const float*)d_in[5];
  const int*   check = (const int*)d_in[6];
  float* out = (float*)d_out;
  float* h   = (float*)d_ws;   // 8*128*56*56 floats = 12.85 MB scratch

  // Output packing (reference tuple, flat, in return order):
  const int OFF_SCORE = 0;                       // (8,256,54,54) = 5,971,968
  const int OFF_IDM   = 5971968;                 // (N,2,3)       =   139,968
  const int OFF_OUT   = 6111936;                 // (8,6,54,54)   =   139,968
  const int OFF_AUXX  = 6251904;                 // (N,3,3)       =   209,952
  const int OFF_AUXY  = 6461856;                 // (N,3,3)       =   209,952

  // K1: conv0 + bias + relu -> h
  conv0_kernel<<<dim3(4, HH, BB), 256, 0, stream>>>(x, w0, b0, h);

  // K2: conv1 * (1-check) + ident -> out6 (also ident23 - theta)
  const int tot2 = BB * 6 * HO * WO;
  conv1_kernel<<<(tot2 + 255) / 256, 256, 0, stream>>>(
      h, w1, check, out + OFF_OUT, out + OFF_IDM);

  // K3: fused grid + sample + GEMM + score scatter (+ auxx/auxy)
  const size_t smem = (size_t)(16 * SROW + 288) * sizeof(float) +
                      (size_t)(144 * 2 + 16) * sizeof(int);
  sample_gemm_kernel<<<NTOT / 16, 256, smem, stream>>>(
      x, out + OFF_OUT, lw, lb, out + OFF_SCORE, out + OFF_AUXX, out + OFF_AUXY);
}